// AnomalyGraph_80487687127505
// MI455X (gfx1250) — compile-verified
//
#include <hip/hip_runtime.h>
#include <math.h>

#define NB   8
#define WIN  256
#define NF   512
#define HID  64
#define LN_EPS 1e-5f

typedef float v2f __attribute__((ext_vector_type(2)));
typedef float v8f __attribute__((ext_vector_type(8)));

// D = A(16x4, f32) * B(4x16, f32) + C(16x16, f32)
__device__ __forceinline__ v8f wmma4(v2f a, v2f b, v8f c) {
    return __builtin_amdgcn_wmma_f32_16x16x4_f32(
        /*neg_a=*/false, a, /*neg_b=*/false, b,
        /*c_mod=*/(short)0, c, /*reuse_a=*/false, /*reuse_b=*/false);
}

// ---- block reductions (256 threads == 8 wave32) -------------------------
__device__ __forceinline__ float blk_sum256(float v, float* red) {
    for (int off = 16; off > 0; off >>= 1) v += __shfl_xor(v, off, 32);
    int wid = threadIdx.x >> 5, ln = threadIdx.x & 31;
    __syncthreads();
    if (ln == 0) red[wid] = v;
    __syncthreads();
    float s = 0.f;
    #pragma unroll
    for (int w = 0; w < 8; ++w) s += red[w];
    return s;
}

__device__ __forceinline__ float blk_max256(float v, float* red) {
    for (int off = 16; off > 0; off >>= 1) v = fmaxf(v, __shfl_xor(v, off, 32));
    int wid = threadIdx.x >> 5, ln = threadIdx.x & 31;
    __syncthreads();
    if (ln == 0) red[wid] = v;
    __syncthreads();
    float m = red[0];
    #pragma unroll
    for (int w = 1; w < 8; ++w) m = fmaxf(m, red[w]);
    return m;
}

// =========================================================================
// K1: node = x^T @ W_fp^T + b_fp ; hi = node @ W_e1[:,:H]^T ; hj (transposed)
// grid: NB*(NF/16) blocks, 128 threads (4 waves; wave w -> H-tile 16w)
// =========================================================================
__global__ void k_node(const float* __restrict__ x, const float* __restrict__ W_fp,
                       const float* __restrict__ b_fp, const float* __restrict__ W_e1,
                       float* __restrict__ node, float* __restrict__ hi,
                       float* __restrict__ hjT)
{
    const int blk   = blockIdx.x;
    const int b     = blk / (NF / 16);
    const int fbase = (blk % (NF / 16)) * 16;
    const int lane  = threadIdx.x & 31;
    const int wave  = threadIdx.x >> 5;
    const int hbase = wave * 16;
    const int row   = lane & 15;            // m for A / n for B,C,D
    const int khalf = (lane >> 4) * 2;      // 0 or 2
    const int moff  = (lane >= 16) ? 8 : 0; // C/D row offset

    __shared__ float tile[16 * 65];         // node tile, padded

    // ---- node tile: A = x^T[b] (512x256 strided), B = W_fp^T (256x64) ----
    const float* xb = x + (size_t)b * WIN * NF;
    v8f acc = {};
    for (int k0 = 0; k0 < WIN; k0 += 4) {
        v2f a, bb;
        a.x  = xb[(size_t)(k0 + khalf)     * NF + (fbase + row)];
        a.y  = xb[(size_t)(k0 + khalf + 1) * NF + (fbase + row)];
        bb.x = W_fp[(size_t)(hbase + row) * WIN + (k0 + khalf)];
        bb.y = W_fp[(size_t)(hbase + row) * WIN + (k0 + khalf + 1)];
        acc = wmma4(a, bb, acc);
    }
    const float bias = b_fp[hbase + row];   // depends on n only
    #pragma unroll
    for (int v = 0; v < 8; ++v) {
        int m = v + moff;
        float val = acc[v] + bias;
        node[((size_t)b * NF + fbase + m) * HID + hbase + row] = val;
        tile[m * 65 + hbase + row] = val;
    }
    __syncthreads();

    // ---- hi / hj tiles: A = node tile (16x64 in LDS), K = 64 ----
    v8f ai = {}, aj = {};
    for (int k0 = 0; k0 < HID; k0 += 4) {
        v2f a, bi, bj;
        a.x = tile[row * 65 + k0 + khalf];
        a.y = tile[row * 65 + k0 + khalf + 1];
        const float* we = W_e1 + (size_t)(hbase + row) * (2 * HID);
        bi.x = we[k0 + khalf];
        bi.y = we[k0 + khalf + 1];
        bj.x = we[HID + k0 + khalf];
        bj.y = we[HID + k0 + khalf + 1];
        ai = wmma4(a, bi, ai);
        aj = wmma4(a, bj, aj);
    }
    #pragma unroll
    for (int v = 0; v < 8; ++v) {
        int m = v + moff;
        hi [((size_t)b * NF  + fbase + m) * HID + hbase + row] = ai[v];
        hjT[((size_t)b * HID + hbase + row) * NF + fbase + m]  = aj[v];  // [h][f]
    }
}

// =========================================================================
// K2: scores (relu-dot, VALU) + diag mask + softmax -> edge_w
// grid: NB*NF blocks (one per (b,i)), 256 threads (2 j's each)
// =========================================================================
__global__ void k_attn(const float* __restrict__ hi, const float* __restrict__ hjT,
                       const float* __restrict__ b_e1, const float* __restrict__ W_e2,
                       const float* __restrict__ b_e2, float* __restrict__ edge_w)
{
    const int blk = blockIdx.x;
    const int b   = blk / NF;
    const int i   = blk % NF;
    const int tid = threadIdx.x;

    __shared__ float hrow[HID];
    __shared__ float wrow[HID];
    __shared__ float brow[HID];
    __shared__ float red[8];

    if (tid < HID) {
        hrow[tid] = hi[((size_t)b * NF + i) * HID + tid];
        wrow[tid] = W_e2[tid];
        brow[tid] = b_e1[tid];
    }
    __syncthreads();

    const float* hj_b = hjT + (size_t)b * HID * NF;
    const float  be2  = b_e2[0];

    float sc[2];
    #pragma unroll
    for (int t = 0; t < 2; ++t) {
        const int j = tid + t * 256;
        float s = be2;
        #pragma unroll 4
        for (int h = 0; h < HID; ++h) {
            float v = hrow[h] + hj_b[(size_t)h * NF + j] + brow[h];
            s = fmaf(fmaxf(v, 0.f), wrow[h], s);
        }
        sc[t] = (j == i) ? -INFINITY : s;
    }

    // softmax over 512 values
    float m = blk_max256(fmaxf(sc[0], sc[1]), red);
    float e0 = __expf(sc[0] - m);
    float e1 = __expf(sc[1] - m);
    float ssum = blk_sum256(e0 + e1, red);
    float inv = 1.f / ssum;

    float* erow = edge_w + ((size_t)b * NF + i) * NF;
    erow[tid]       = e0 * inv;
    erow[tid + 256] = e1 * inv;
}

// =========================================================================
// K3: edge_w_sum[b,j] = sum_i edge_w[b,i,j]
// grid: NB*(NF/256) blocks, 256 threads (one j per thread, coalesced)
// =========================================================================
__global__ void k_ewsum(const float* __restrict__ edge_w, float* __restrict__ ews)
{
    const int b = blockIdx.x / (NF / 256);
    const int j = (blockIdx.x % (NF / 256)) * 256 + threadIdx.x;
    const float* e = edge_w + (size_t)b * NF * NF + j;
    float s = 0.f;
    for (int i = 0; i < NF; ++i) s += e[(size_t)i * NF];
    ews[b * NF + j] = s;
}

// =========================================================================
// K4: messages = edge_w @ node   (K = 512)
// grid: NB*(NF/16) blocks, 128 threads (4 waves -> HID tiles)
// =========================================================================
__global__ void k_msg(const float* __restrict__ edge_w, const float* __restrict__ node,
                      float* __restrict__ messages)
{
    const int blk   = blockIdx.x;
    const int b     = blk / (NF / 16);
    const int fbase = (blk % (NF / 16)) * 16;
    const int lane  = threadIdx.x & 31;
    const int wave  = threadIdx.x >> 5;
    const int hbase = wave * 16;
    const int row   = lane & 15;
    const int khalf = (lane >> 4) * 2;
    const int moff  = (lane >= 16) ? 8 : 0;

    const float* A  = edge_w + (size_t)b * NF * NF;
    const float* Bm = node   + (size_t)b * NF * HID;

    v8f acc = {};
    for (int k0 = 0; k0 < NF; k0 += 4) {
        v2f a, bb;
        a.x  = A[(size_t)(fbase + row) * NF + k0 + khalf];
        a.y  = A[(size_t)(fbase + row) * NF + k0 + khalf + 1];
        bb.x = Bm[(size_t)(k0 + khalf)     * HID + hbase + row];
        bb.y = Bm[(size_t)(k0 + khalf + 1) * HID + hbase + row];
        acc = wmma4(a, bb, acc);
    }
    #pragma unroll
    for (int v = 0; v < 8; ++v) {
        int m = v + moff;
        messages[((size_t)b * NF + fbase + m) * HID + hbase + row] = acc[v];
    }
}

// =========================================================================
// K5: pre[b,w,f] = (messages @ W_op^T)[f,w] + b_op[w] + x[b,w,f]   (K = 64)
// grid: NB*(NF/16)*(WIN/64) blocks, 128 threads (4 waves -> WIN tiles)
// =========================================================================
__global__ void k_outfeat(const float* __restrict__ messages, const float* __restrict__ W_op,
                          const float* __restrict__ b_op, const float* __restrict__ x,
                          float* __restrict__ pre)
{
    const int blk    = blockIdx.x;
    const int wchunk = blk % (WIN / 64);
    const int tmp    = blk / (WIN / 64);
    const int fbase  = (tmp % (NF / 16)) * 16;
    const int b      = tmp / (NF / 16);
    const int lane   = threadIdx.x & 31;
    const int wave   = threadIdx.x >> 5;
    const int wbase  = wchunk * 64 + wave * 16;
    const int row    = lane & 15;
    const int khalf  = (lane >> 4) * 2;
    const int moff   = (lane >= 16) ? 8 : 0;

    const float* A = messages + ((size_t)b * NF + fbase) * HID;

    v8f acc = {};
    for (int k0 = 0; k0 < HID; k0 += 4) {
        v2f a, bb;
        a.x  = A[(size_t)row * HID + k0 + khalf];
        a.y  = A[(size_t)row * HID + k0 + khalf + 1];
        bb.x = W_op[(size_t)(wbase + row) * HID + k0 + khalf];
        bb.y = W_op[(size_t)(wbase + row) * HID + k0 + khalf + 1];
        acc = wmma4(a, bb, acc);
    }
    const float bias = b_op[wbase + row];   // n -> w
    #pragma unroll
    for (int v = 0; v < 8; ++v) {
        int m = v + moff;                   // m -> f
        size_t idx = ((size_t)b * WIN + wbase + row) * NF + fbase + m;
        pre[idx] = acc[v] + bias + x[idx];
    }
}

// =========================================================================
// K6: LayerNorm over F per (b,w); write out + ew_expanded
// grid: NB*WIN blocks, 256 threads (2 f's each)
// =========================================================================
__global__ void k_ln(const float* __restrict__ pre, const float* __restrict__ gamma,
                     const float* __restrict__ beta, const float* __restrict__ ews,
                     float* __restrict__ out, float* __restrict__ ew_out)
{
    const int blk = blockIdx.x;
    const int b   = blk / WIN;
    const int w   = blk % WIN;
    const int tid = threadIdx.x;

    __shared__ float red[8];

    const size_t base = ((size_t)b * WIN + w) * NF;
    const float v0 = pre[base + tid];
    const float v1 = pre[base + tid + 256];

    float s = blk_sum256(v0 + v1, red);
    const float mu = s * (1.f / NF);
    const float d0 = v0 - mu, d1 = v1 - mu;
    float q = blk_sum256(d0 * d0 + d1 * d1, red);
    const float rstd = rsqrtf(q * (1.f / NF) + LN_EPS);

    out[base + tid]        = d0 * rstd * gamma[tid]       + beta[tid];
    out[base + tid + 256]  = d1 * rstd * gamma[tid + 256] + beta[tid + 256];
    ew_out[base + tid]       = ews[b * NF + tid];
    ew_out[base + tid + 256] = ews[b * NF + tid + 256];
}

// =========================================================================
extern "C" void kernel_launch(void* const* d_in, const int* in_sizes, int n_in,
                              void* d_out, int out_size, void* d_ws, size_t ws_size,
                              hipStream_t stream) {
    const float* x     = (const float*)d_in[0];
    const float* W_fp  = (const float*)d_in[1];
    const float* b_fp  = (const float*)d_in[2];
    const float* W_e1  = (const float*)d_in[3];
    const float* b_e1  = (const float*)d_in[4];
    const float* W_e2  = (const float*)d_in[5];
    const float* b_e2  = (const float*)d_in[6];
    const float* W_op  = (const float*)d_in[7];
    const float* b_op  = (const float*)d_in[8];
    const float* gamma = (const float*)d_in[9];
    const float* beta  = (const float*)d_in[10];

    float* out    = (float*)d_out;                       // [B, WIN, NF]
    float* ew_out = out + (size_t)NB * WIN * NF;         // [B, WIN, NF]

    // workspace carve-up (floats)
    float* ws       = (float*)d_ws;
    float* node     = ws;                                    // B*NF*HID
    float* hi       = node     + (size_t)NB * NF * HID;      // B*NF*HID
    float* hjT      = hi       + (size_t)NB * NF * HID;      // B*HID*NF
    float* edge_w   = hjT      + (size_t)NB * HID * NF;      // B*NF*NF
    float* ews      = edge_w   + (size_t)NB * NF * NF;       // B*NF
    float* messages = ews      + (size_t)NB * NF;            // B*NF*HID
    float* pre      = messages + (size_t)NB * NF * HID;      // B*WIN*NF

    k_node   <<<NB * (NF / 16),               128, 0, stream>>>(x, W_fp, b_fp, W_e1, node, hi, hjT);
    k_attn   <<<NB * NF,                      256, 0, stream>>>(hi, hjT, b_e1, W_e2, b_e2, edge_w);
    k_ewsum  <<<NB * (NF / 256),              256, 0, stream>>>(edge_w, ews);
    k_msg    <<<NB * (NF / 16),               128, 0, stream>>>(edge_w, node, messages);
    k_outfeat<<<NB * (NF / 16) * (WIN / 64),  128, 0, stream>>>(messages, W_op, b_op, x, pre);
    k_ln     <<<NB * WIN,                     256, 0, stream>>>(pre, gamma, beta, ews, out, ew_out);
}